// SSDLayer_62637803045608
// MI455X (gfx1250) — compile-verified
//
#include <hip/hip_runtime.h>
#include <math.h>
#include <stdint.h>

#define BDIM   256
#define NWAVES (BDIM / 32)
#define M_GT   32
#define NCLS   21

typedef float v2f __attribute__((ext_vector_type(2)));
typedef float v8f __attribute__((ext_vector_type(8)));

// Sum the 32 per-lane floats of a wave with one V_WMMA_F32_16X16X4_F32:
// D = ones(16x4) x B(4x16) + 0  =>  D[m][n] = column-sum of B.
// The 32 partials fill exactly half the 64 B slots (b[1]=0 zeroes the rest),
// so the 16 column sums add to the wave total. c[0] across lanes 0..31 holds
// the 16 column sums twice; a 32-lane butterfly therefore yields 2x total.
__device__ __forceinline__ float wave_reduce_sum_wmma(float x) {
  v2f a; a[0] = 1.0f; a[1] = 1.0f;
  v2f b; b[0] = x;    b[1] = 0.0f;
  v8f c = {};
  c = __builtin_amdgcn_wmma_f32_16x16x4_f32(false, a, false, b, (short)0, c,
                                            false, false);
  float s = c[0];
#pragma unroll
  for (int off = 16; off > 0; off >>= 1)
    s += __shfl_xor(s, off, 32);
  return 0.5f * s;
}

__device__ __forceinline__ float smooth_l1(float d) {
  float a = fabsf(d);
  return (a < 1.0f) ? 0.5f * d * d : a - 0.5f;
}

__global__ void k_init(int* npos) { *npos = 0; }

// One block per batch row: matching + smooth-L1 + CE in a single pass over p.
__global__ __launch_bounds__(BDIM)
void k_match(const float* __restrict__ p, const float* __restrict__ tg,
             const float* __restrict__ pri, int P,
             float* __restrict__ ce_out, unsigned* __restrict__ key_out,
             unsigned char* __restrict__ pos_out,
             float* __restrict__ blk_sl1, int* __restrict__ npos_g)
{
  __shared__ float gx1[M_GT], gy1[M_GT], gx2[M_GT], gy2[M_GT];
  __shared__ float garea[M_GT], glab[M_GT];
  __shared__ int   gvalid[M_GT];
  __shared__ unsigned long long bestpk[M_GT];
  __shared__ unsigned bp[M_GT];
  __shared__ float wsl1[NWAVES], wcnt[NWAVES];

  const int b   = blockIdx.x;
  const int tid = threadIdx.x;

  if (tid < M_GT) {
    const float* t = tg + ((size_t)b * M_GT + tid) * 6;
    gvalid[tid] = (t[0] > 0.0f);
    glab[tid]   = t[1];
    gx1[tid] = t[2]; gy1[tid] = t[3]; gx2[tid] = t[4]; gy2[tid] = t[5];
    garea[tid]  = (t[4] - t[2]) * (t[5] - t[3]);
    bestpk[tid] = 0ull;
  }
  __syncthreads();

  // Pass 1: per-GT argmax over priors via packed LDS atomicMax.
  // Key = (iou bits << 32) | (~j)  -> ties resolve to the lowest prior index,
  // matching jnp.argmax; invalid GTs never scatter (forced stays -1).
  for (int j = tid; j < P; j += BDIM) {
    float cx = pri[4*j+0], cy = pri[4*j+1], w = pri[4*j+2], h = pri[4*j+3];
    float px1 = cx - 0.5f*w, py1 = cy - 0.5f*h;
    float px2 = cx + 0.5f*w, py2 = cy + 0.5f*h;
    float pa  = (px2 - px1) * (py2 - py1);
    for (int m = 0; m < M_GT; ++m) {
      if (!gvalid[m]) continue;
      float iw = fmaxf(fminf(gx2[m], px2) - fmaxf(gx1[m], px1), 0.0f);
      float ih = fmaxf(fminf(gy2[m], py2) - fmaxf(gy1[m], py1), 0.0f);
      float inter = iw * ih;
      float iou = inter / (garea[m] + pa - inter);
      unsigned long long pk =
          ((unsigned long long)__float_as_uint(iou) << 32) |
          (unsigned long long)(0xFFFFFFFFu - (unsigned)j);
      atomicMax(&bestpk[m], pk);
    }
  }
  __syncthreads();
  if (tid < M_GT)
    bp[tid] = gvalid[tid] ? (0xFFFFFFFFu - (unsigned)(bestpk[tid] & 0xFFFFFFFFull))
                          : 0xFFFFFFFFu;
  __syncthreads();

  // Pass 2: per-prior best GT (+forced override), loc + conf losses.
  float sl1_part = 0.0f, cnt_part = 0.0f;
  for (int j = tid; j < P; j += BDIM) {
    float cx = pri[4*j+0], cy = pri[4*j+1], w = pri[4*j+2], h = pri[4*j+3];
    float px1 = cx - 0.5f*w, py1 = cy - 0.5f*h;
    float px2 = cx + 0.5f*w, py2 = cy + 0.5f*h;
    float pa  = (px2 - px1) * (py2 - py1);

    float bestiou = -2.0f; int besti = 0;
    for (int m = 0; m < M_GT; ++m) {
      float iou = -1.0f;                      // reference: invalid rows = -1
      if (gvalid[m]) {
        float iw = fmaxf(fminf(gx2[m], px2) - fmaxf(gx1[m], px1), 0.0f);
        float ih = fmaxf(fminf(gy2[m], py2) - fmaxf(gy1[m], py1), 0.0f);
        float inter = iw * ih;
        iou = inter / (garea[m] + pa - inter);
      }
      if (iou > bestiou) { bestiou = iou; besti = m; }  // first-max wins
    }
    int forced = -1;
    for (int m = 0; m < M_GT; ++m)
      if (gvalid[m] && bp[m] == (unsigned)j) forced = m; // ascending -> max m
    if (forced >= 0) { besti = forced; bestiou = 2.0f; }

    float conf = (bestiou < 0.5f) ? 0.0f : glab[besti];
    int pos = (conf > 0.0f);

    const float* row = p + ((size_t)b * P + j) * (4 + NCLS);
    if (j + BDIM < P)
      __builtin_prefetch(row + (size_t)BDIM * (4 + NCLS), 0, 3);

    // localization target + smooth-L1 (summed only over positives)
    float gcx = 0.5f * (gx1[besti] + gx2[besti]);
    float gcy = 0.5f * (gy1[besti] + gy2[besti]);
    float gw  = gx2[besti] - gx1[besti];
    float gh  = gy2[besti] - gy1[besti];
    float lt0 = gcx / w - 1.0f;
    float lt1 = gcy / h - 1.0f;
    float lt2 = logf(fmaxf(gw, 1e-8f) / w);
    float lt3 = logf(fmaxf(gh, 1e-8f) / h);
    float s = smooth_l1(row[0] - lt0) + smooth_l1(row[1] - lt1) +
              smooth_l1(row[2] - lt2) + smooth_l1(row[3] - lt3);
    if (pos) { sl1_part += s; cnt_part += 1.0f; }

    // cross-entropy: logsumexp over 21 logits minus true logit
    float mx = row[4];
#pragma unroll
    for (int k = 1; k < NCLS; ++k) mx = fmaxf(mx, row[4 + k]);
    float se = 0.0f;
#pragma unroll
    for (int k = 0; k < NCLS; ++k) se += expf(row[4 + k] - mx);
    float cev = (mx + logf(se)) - row[4 + (int)conf];

    size_t o = (size_t)b * P + j;
    ce_out[o]  = cev;
    key_out[o] = __float_as_uint(pos ? 0.0f : cev);  // mining key (>= 0)
    pos_out[o] = (unsigned char)pos;
  }

  float wsum = wave_reduce_sum_wmma(sl1_part);
  float wpos = wave_reduce_sum_wmma(cnt_part);
  if ((tid & 31) == 0) { wsl1[tid >> 5] = wsum; wcnt[tid >> 5] = wpos; }
  __syncthreads();
  if (tid == 0) {
    float t1 = 0.0f, t2 = 0.0f;
    for (int i = 0; i < NWAVES; ++i) { t1 += wsl1[i]; t2 += wcnt[i]; }
    blk_sl1[b] = t1;
    atomicAdd(npos_g, (int)(t2 + 0.5f));   // int atomics: deterministic
  }
}

// Hard-negative mining without sorting: binary search on float-key space for
// the num_neg-th largest mining value, then stable (index-ordered) tie take.
__global__ __launch_bounds__(BDIM)
void k_mine(const float* __restrict__ ce, const unsigned* __restrict__ key,
            const unsigned char* __restrict__ pos, int P,
            const int* __restrict__ npos_g,
            float* __restrict__ blk_ce, float* __restrict__ blk_cnt)
{
  __shared__ int s_cnt;
  __shared__ float wce[NWAVES], wcn[NWAVES];
  const int b = blockIdx.x, tid = threadIdx.x;
  const unsigned*      krow = key + (size_t)b * P;
  const float*         crow = ce  + (size_t)b * P;
  const unsigned char* prow = pos + (size_t)b * P;

  int np = *npos_g;                           // global across the batch
  long long nn  = 3LL * (long long)np;
  long long cap = (long long)P - (long long)np;
  if (nn > cap) nn = cap;
  if (nn < 0)   nn = 0;
  const int num_neg = (int)nn;

  unsigned Tkey = 0xFFFFFFFFu;
  int r = 0;
  if (num_neg > 0) {
    // largest t with count(key >= t) >= num_neg  (keys are IEEE bits of >=0)
    unsigned lo = 0u, hi = 0xFFFFFFFFu;
    while (lo < hi) {
      unsigned mid = (unsigned)(((unsigned long long)lo +
                                 (unsigned long long)hi + 1ull) >> 1);
      if (tid == 0) s_cnt = 0;
      __syncthreads();
      int c = 0;
      for (int j = tid; j < P; j += BDIM) c += (krow[j] >= mid) ? 1 : 0;
      atomicAdd(&s_cnt, c);
      __syncthreads();
      int cnt = s_cnt;
      __syncthreads();
      if (cnt >= num_neg) lo = mid; else hi = mid - 1;
    }
    Tkey = lo;
    if (tid == 0) s_cnt = 0;
    __syncthreads();
    int c = 0;
    for (int j = tid; j < P; j += BDIM) c += (krow[j] > Tkey) ? 1 : 0;
    atomicAdd(&s_cnt, c);
    __syncthreads();
    r = num_neg - s_cnt;                      // slots left at the threshold
    __syncthreads();
  }

  // mask = pos | (rank < num_neg); strictly-above-threshold part here
  float ces = 0.0f, cns = 0.0f;
  for (int j = tid; j < P; j += BDIM) {
    bool inc = prow[j] || (num_neg > 0 && krow[j] > Tkey);
    if (inc) { ces += crow[j]; cns += 1.0f; }
  }
  float a  = wave_reduce_sum_wmma(ces);
  float c2 = wave_reduce_sum_wmma(cns);
  if ((tid & 31) == 0) { wce[tid >> 5] = a; wcn[tid >> 5] = c2; }
  __syncthreads();
  if (tid == 0) {
    float t1 = 0.0f, t2 = 0.0f;
    for (int i = 0; i < NWAVES; ++i) { t1 += wce[i]; t2 += wcn[i]; }
    // stable-sort tie semantics: first r equal-key elements by index consume
    // rank slots; positives among them are already in the mask union.
    int taken = 0;
    for (int j = 0; j < P && taken < r; ++j) {
      if (krow[j] == Tkey) {
        if (!prow[j]) { t1 += crow[j]; t2 += 1.0f; }
        ++taken;
      }
    }
    blk_ce[b]  = t1;
    blk_cnt[b] = t2;
  }
}

__global__ void k_final(const float* __restrict__ blk_sl1,
                        const float* __restrict__ blk_ce,
                        const float* __restrict__ blk_cnt,
                        const int* __restrict__ npos_g,
                        int B, float* __restrict__ out)
{
  float sl1 = 0.0f, cesum = 0.0f, mcnt = 0.0f;
  for (int i = 0; i < B; ++i) {          // fixed order -> deterministic
    sl1 += blk_sl1[i]; cesum += blk_ce[i]; mcnt += blk_cnt[i];
  }
  int np = *npos_g;
  long long dl = 4LL * (long long)np; if (dl < 1) dl = 1;
  float dc = (mcnt > 1.0f) ? mcnt : 1.0f;
  out[0] = sl1 / (float)dl + cesum / dc;
}

extern "C" void kernel_launch(void* const* d_in, const int* in_sizes, int n_in,
                              void* d_out, int out_size, void* d_ws, size_t ws_size,
                              hipStream_t stream) {
  const float* p   = (const float*)d_in[0];   // (B, P, 25)
  const float* tg  = (const float*)d_in[1];   // (B, 32, 6)
  const float* pri = (const float*)d_in[2];   // (P, 4)
  const int P = in_sizes[2] / 4;
  const int B = in_sizes[1] / (M_GT * 6);

  char* w = (char*)d_ws;
  float*         ce  = (float*)w;          w += (size_t)B * P * sizeof(float);
  unsigned*      key = (unsigned*)w;       w += (size_t)B * P * sizeof(unsigned);
  unsigned char* pos = (unsigned char*)w;  w += (size_t)B * P;
  uintptr_t al = ((uintptr_t)w + 15) & ~(uintptr_t)15;
  float* blk_sl1 = (float*)al;
  float* blk_ce  = blk_sl1 + B;
  float* blk_cnt = blk_ce + B;
  int*   npos    = (int*)(blk_cnt + B);

  k_init <<<1, 1, 0, stream>>>(npos);
  k_match<<<B, BDIM, 0, stream>>>(p, tg, pri, P, ce, key, pos, blk_sl1, npos);
  k_mine <<<B, BDIM, 0, stream>>>(ce, key, pos, P, npos, blk_ce, blk_cnt);
  k_final<<<1, 1, 0, stream>>>(blk_sl1, blk_ce, blk_cnt, npos, B, (float*)d_out);
  (void)n_in; (void)out_size; (void)ws_size;
}